// RouterMoEClass_22995254902986
// MI455X (gfx1250) — compile-verified
//
#include <hip/hip_runtime.h>

typedef __attribute__((ext_vector_type(2))) float v2f;
typedef __attribute__((ext_vector_type(8))) float v8f;
typedef __attribute__((ext_vector_type(4))) int   v4i;

#define HDIM 4096
#define EDIM 64
#define KC   64      // K chunk per LDS stage
#define KCP  68      // padded LDS row stride (floats) -> conflict-free ds_load_b64
#define TOKS 128     // tokens per workgroup
#define NT   4       // 64 experts / 16 = 4 WMMA N-tiles

typedef __attribute__((address_space(1))) v4i gv4i;   // global, 16B granule
typedef __attribute__((address_space(3))) v4i lv4i;   // LDS, 16B granule

// One-instruction async 16B copy: copies from global address g to LDS address l, tracked by ASYNCcnt.
__device__ __forceinline__ void async16(const float* g, float* l) {
    __builtin_amdgcn_global_load_async_to_lds_b128(
        (gv4i*)(g), (lv4i*)(l), /*imm offset=*/0, /*cpol=*/0);
}

__device__ __forceinline__ void wait_async_zero() {
    asm volatile("s_wait_asynccnt 0" ::: "memory");
}

// Cooperative stage of one K-chunk: A = 128 tokens x 64 k, B = 64 experts x 64 k.
// 16 lanes cover one contiguous 256B row per async pass.
__device__ __forceinline__ void stage_chunk(const float* __restrict__ hidden,
                                            const float* __restrict__ weight,
                                            float* __restrict__ A,
                                            float* __restrict__ Bq,
                                            int tokBase, int k0, int row0, int col) {
    #pragma unroll
    for (int p = 0; p < 8; ++p) {
        const int row = p * 16 + row0;
        async16(hidden + (size_t)(tokBase + row) * HDIM + k0 + col, &A[row * KCP + col]);
    }
    #pragma unroll
    for (int p = 0; p < 4; ++p) {
        const int row = p * 16 + row0;
        async16(weight + (size_t)row * HDIM + k0 + col, &Bq[row * KCP + col]);
    }
}

__global__ __launch_bounds__(256, 1)
void router_moe_kernel(const float* __restrict__ hidden,
                       const float* __restrict__ weight,
                       float* __restrict__ logits_out,
                       float* __restrict__ affin_out,
                       int*   __restrict__ index_out)
{
    // Double-buffered stages (~102 KB total; LDS is 320 KB/WGP on CDNA5).
    __shared__ float ldsA[2][TOKS * KCP];
    __shared__ float ldsB[2][EDIM * KCP];

    const int tid  = threadIdx.x;
    const int lane = tid & 31;
    const int wave = tid >> 5;
    const int tokBase = blockIdx.x * TOKS;

    const int m16   = lane & 15;          // M (A) / N (B) index within tile
    const int khalf = (lane >> 4) << 1;   // lanes 0-15 -> K 0,1 ; lanes 16-31 -> K 2,3

    const int row0 = tid >> 4;            // 0..15 : cooperative-load row group
    const int col  = (tid & 15) << 2;     // 0..60 : float4 column

    v8f acc[NT];
    #pragma unroll
    for (int t = 0; t < NT; ++t) acc[t] = v8f{0.f,0.f,0.f,0.f,0.f,0.f,0.f,0.f};

    // Prologue: stage chunk 0.
    stage_chunk(hidden, weight, ldsA[0], ldsB[0], tokBase, 0, row0, col);
    wait_async_zero();
    __syncthreads();

    const int nchunks = HDIM / KC;        // 64
    for (int c = 0; c < nchunks; ++c) {
        const int cur = c & 1;

        // Prefetch next chunk into the alternate buffer while computing.
        if (c + 1 < nchunks)
            stage_chunk(hidden, weight, ldsA[cur ^ 1], ldsB[cur ^ 1],
                        tokBase, (c + 1) * KC, row0, col);

        const float* aRow  = &ldsA[cur][(wave * 16 + m16) * KCP + khalf];
        const float* bBase = &ldsB[cur][m16 * KCP + khalf];

        // 16 K-steps x 4 N-tiles = 64 v_wmma_f32_16x16x4_f32 per chunk.
        #pragma unroll
        for (int kk = 0; kk < KC; kk += 4) {
            v2f a = *(const v2f*)(aRow + kk);
            #pragma unroll
            for (int t = 0; t < NT; ++t) {
                v2f b = *(const v2f*)(bBase + t * 16 * KCP + kk);
                acc[t] = __builtin_amdgcn_wmma_f32_16x16x4_f32(
                    /*neg_a=*/false, a, /*neg_b=*/false, b,
                    /*c_mod=*/(short)0, acc[t],
                    /*reuse_a=*/false, /*reuse_b=*/false);
            }
        }

        wait_async_zero();   // prefetched buffer fully landed in LDS
        __syncthreads();     // all waves done reading current buffer
    }

    // ---- epilogue: spill logits to global + LDS (reuse buffer 0's A region) ----
    float* lgrid = &ldsA[0][0];
    const int mhalf = (lane >> 4) << 3;   // 0 or 8
    #pragma unroll
    for (int t = 0; t < NT; ++t) {
        #pragma unroll
        for (int r = 0; r < 8; ++r) {
            const int tl = wave * 16 + mhalf + r;   // local token
            const int e  = t * 16 + m16;            // expert index
            const float v = acc[t][r];
            logits_out[(size_t)(tokBase + tl) * EDIM + e] = v;
            lgrid[tl * KCP + e] = v;
        }
    }
    __syncthreads();

    // ---- softmax + top-2 : one thread per token ----
    if (tid < TOKS) {
        const float* row = &lgrid[tid * KCP];
        float mx = row[0];
        #pragma unroll 8
        for (int e = 1; e < EDIM; ++e) mx = fmaxf(mx, row[e]);
        float sum = 0.f;
        #pragma unroll 8
        for (int e = 0; e < EDIM; ++e) sum += __expf(row[e] - mx);
        const float inv = 1.f / sum;

        float b1 = -INFINITY, b2 = -INFINITY;
        int   i1 = 0,         i2 = 0;
        #pragma unroll 8
        for (int e = 0; e < EDIM; ++e) {
            const float lg  = row[e];
            const float aff = __expf(lg - mx) * inv;
            affin_out[(size_t)(tokBase + tid) * EDIM + e] = aff;
            if (lg > b1)      { b2 = b1; i2 = i1; b1 = lg; i1 = e; }
            else if (lg > b2) { b2 = lg; i2 = e; }
        }
        index_out[(size_t)(tokBase + tid) * 2 + 0] = i1;
        index_out[(size_t)(tokBase + tid) * 2 + 1] = i2;
    }
}

extern "C" void kernel_launch(void* const* d_in, const int* in_sizes, int n_in,
                              void* d_out, int out_size, void* d_ws, size_t ws_size,
                              hipStream_t stream) {
    const float* hidden = (const float*)d_in[0];   // [B,S,H] fp32
    const float* weight = (const float*)d_in[1];   // [E,H]   fp32

    const long nTok = (long)in_sizes[0] / HDIM;    // B*S = 32768
    float* out    = (float*)d_out;
    float* logits = out;                                   // [nTok, 64]
    float* affin  = out + (size_t)nTok * EDIM;             // [nTok, 64]
    int*   idx    = (int*)(out + 2 * (size_t)nTok * EDIM); // [nTok, 2] int32

    dim3 grid((unsigned)(nTok / TOKS));
    router_moe_kernel<<<grid, 256, 0, stream>>>(hidden, weight, logits, affin, idx);
}